// MA_16192026706454
// MI455X (gfx1250) — compile-verified
//
#include <hip/hip_runtime.h>

typedef __attribute__((ext_vector_type(2))) float v2f;
typedef __attribute__((ext_vector_type(8))) float v8f;

#define XST 608                 // LDS row stride for Xh (600 cols + pad cols + slack)
#define XLS_TOT (64 * XST + 80) // slack absorbs guarded reads from padded m / o columns

__global__ __launch_bounds__(256, 1)
void fused_conv_bmm(const float* __restrict__ X,   // [1024][64][600]  (B*H, d_k, L)
                    const float* __restrict__ Wg,  // [600][64][3]
                    const float* __restrict__ Bb,  // [600]
                    float* __restrict__ Out)       // [1024][64][600]
{
    __shared__ float xls[XLS_TOT];     // Xh[n], zero-padded:  xls[i*XST + (l+1)]
    __shared__ float Wls[96 * 32];     // W block, [K/2][o] pair-interleaved (K = t*64+i)
    __shared__ float Cls[128 * 16];    // C block staged as [m_local][o]
    __shared__ float bls[608];         // bias, zero-padded

    const int tid  = threadIdx.x;
    const int lane = tid & 31;
    const int half = lane >> 4;   // 0: lanes 0-15, 1: lanes 16-31
    const int l16  = lane & 15;
    const int wave = tid >> 5;    // 0..7
    const int bh   = blockIdx.x;  // 0..1023
    const int mw   = blockIdx.y * 128 + wave * 16;  // wave's first output column

    // ---- zero LDS (guarantees padded/garbage regions are finite 0, so 0*x stays 0)
    for (int idx = tid; idx < XLS_TOT; idx += 256) xls[idx] = 0.0f;
    for (int idx = tid; idx < 608; idx += 256) bls[idx] = (idx < 600) ? Bb[idx] : 0.0f;
    __syncthreads();

    // ---- load the whole Xh[n] slice (64x600 f32 = 150KB) into LDS
    const float* xsrc = X + (size_t)bh * (64 * 600);
    for (int idx = tid; idx < 64 * 600; idx += 256) {
        int i = idx / 600;
        int l = idx - i * 600;
        xls[i * XST + l + 1] = xsrc[idx];
    }
    __syncthreads();

    v8f acc0 = {}, acc1 = {}, acc2 = {}, acc3 = {};  // 64 x 16 output strip per wave

    for (int ob = 0; ob < 38; ++ob) {            // 608 out-channels in blocks of 16
        const int obase = ob * 16;

        // ---- stage W block -> LDS, transposed to [K][o], K = t*64 + i, pairs interleaved
        for (int idx = tid; idx < 16 * 192; idx += 256) {
            int orow = idx / 192;
            int K    = idx - orow * 192;
            int t = K >> 6, i = K & 63;
            int o = obase + orow;
            float w = (o < 600) ? Wg[o * 192 + i * 3 + t] : 0.0f;
            Wls[(K >> 1) * 32 + orow * 2 + (K & 1)] = w;
        }
        __syncthreads();

        // ---- conv GEMM: Cf(16x16) = Wblk(16x192) @ P(192x16), K-steps of 4
        v8f cf = {};
        for (int ks = 0; ks < 48; ++ks) {
            int k0 = ks * 4;
            int t  = k0 >> 6;       // which conv tap
            int i0 = k0 & 63;       // input-channel base
            // A fragment (W): lane<16 -> K=k0,k0+1 ; lane>=16 -> K=k0+2,k0+3
            const float* wp = &Wls[(2 * ks + half) * 32 + l16 * 2];
            v2f a; a.x = wp[0]; a.y = wp[1];
            // B fragment (P from Xh in LDS): row K -> xh[i, m+t-1] ; array col = m+t
            int irow = i0 + 2 * half;
            int col  = mw + l16 + t;
            v2f bfrag;
            bfrag.x = xls[irow * XST + col];
            bfrag.y = xls[(irow + 1) * XST + col];
            cf = __builtin_amdgcn_wmma_f32_16x16x4_f32(false, a, false, bfrag,
                                                       (short)0, cf, false, false);
        }
        // bias: C/D layout -> element v is row o = obase + v + 8*half
        #pragma unroll
        for (int v = 0; v < 8; ++v) cf[v] += bls[obase + v + 8 * half];

        // ---- stage C tile to LDS as [m_local][o] (layout shuffle D -> B)
        {
            int base = (wave * 16 + l16) * 16 + 8 * half;
            #pragma unroll
            for (int v = 0; v < 8; ++v) Cls[base + v] = cf[v];
        }
        __syncthreads();

        // ---- bmm accumulate: acc(64 x 16) += Xh[:, obase:obase+16] @ Cblk(16 x 16)
        v2f bf[4];
        #pragma unroll
        for (int ks = 0; ks < 4; ++ks) {
            int kk = ks * 4 + 2 * half;
            const float* cp = &Cls[(wave * 16 + l16) * 16 + kk];
            bf[ks].x = cp[0]; bf[ks].y = cp[1];
        }
        #pragma unroll
        for (int kb = 0; kb < 4; ++kb) {
            v8f* accp = (kb == 0) ? &acc0 : (kb == 1) ? &acc1 : (kb == 2) ? &acc2 : &acc3;
            #pragma unroll
            for (int ks = 0; ks < 4; ++ks) {
                int Kl   = ks * 4 + 2 * half;
                int arow = kb * 16 + l16;
                v2f a;
                a.x = xls[arow * XST + 1 + obase + Kl];
                a.y = xls[arow * XST + 2 + obase + Kl];
                *accp = __builtin_amdgcn_wmma_f32_16x16x4_f32(false, a, false, bf[ks],
                                                              (short)0, *accp, false, false);
            }
        }
        __syncthreads();
    }

    // ---- store: out[(bh*64 + k)*600 + m], D layout: m = mw+l16, k = kb*16 + v + 8*half
    int m = mw + l16;
    if (m < 600) {
        float* op = Out + (size_t)bh * 64 * 600 + m;
        #pragma unroll
        for (int kb = 0; kb < 4; ++kb) {
            v8f av = (kb == 0) ? acc0 : (kb == 1) ? acc1 : (kb == 2) ? acc2 : acc3;
            #pragma unroll
            for (int v = 0; v < 8; ++v) {
                int k = kb * 16 + v + 8 * half;
                op[(size_t)k * 600] = av[v];
            }
        }
    }
}

extern "C" void kernel_launch(void* const* d_in, const int* in_sizes, int n_in,
                              void* d_out, int out_size, void* d_ws, size_t ws_size,
                              hipStream_t stream) {
    const float* x = (const float*)d_in[0];   // [64,1024,600] fp32
    const float* W = (const float*)d_in[1];   // [600,64,3]    fp32
    const float* b = (const float*)d_in[2];   // [600]         fp32
    float* out = (float*)d_out;               // [64,1024,600] fp32

    dim3 grid(1024, 5);   // (B*H, m-tiles of 128)
    fused_conv_bmm<<<grid, dim3(256), 0, stream>>>(x, W, b, out);
}